// Tile_36807869726866
// MI455X (gfx1250) — compile-verified
//
#include <hip/hip_runtime.h>
#include <math.h>
#include <stdint.h>

// Problem constants (match reference: K=8,S=4,B=16,C=3,H=W=512)
#define Bb     16
#define Cc     3
#define Hh     512
#define Ww     512
#define NHh    127
#define NWw    127
#define LL     (NHh * NWw)          // 16129
#define EPP    192                  // C*K*K elements per patch
#define CH_J   8                    // patches per j-chunk
#define NCHUNK 16                   // ceil(127/8)
#define REG_COLS 36                 // 4*CH_J + 4 pixel columns per region
#define TPB    384                  // 8 patches * 48 float4-slots

// u0 staged region: patch rows i-1..i+1 (3) x patch cols j0-1..j0+8 (10)
#define NSLOT  30
#define SMU_FLOATS (NSLOT * EPP)            // 5760
#define SMCOEF_FLOATS (Cc * 8 * REG_COLS)   // 864

// ---- CDNA5 async global->LDS copy (tracked by ASYNCcnt) -------------------
__device__ __forceinline__ void async_copy_b128(uint32_t lds_byte_off,
                                                const float* gsrc) {
    uint64_t gaddr = (uint64_t)(uintptr_t)gsrc;
    asm volatile("global_load_async_to_lds_b128 %0, %1, off"
                 :: "v"(lds_byte_off), "v"(gaddr)
                 : "memory");
}

__device__ __forceinline__ void wait_async_all() {
    asm volatile("s_wait_asynccnt 0x0" ::: "memory");
}

// ---------------------------------------------------------------------------
// Fused kernel: stage u0 -> LDS (async), compute per-pixel coefA/coefC in
// LDS, then tiles[b,l,e] = coefA(pix)*u0[b,l,e] + coefC(pix).
// grid = (16 j-chunks, 127 i, 16 b), block = 384.
// ---------------------------------------------------------------------------
__global__ __launch_bounds__(TPB)
void fused_tiles_kernel(const float* __restrict__ ximg,   // (B,C,H,W)
                        const float* __restrict__ u0,     // (B,L,192)
                        const float* __restrict__ rimg,   // (B,C,H,W)
                        const float* __restrict__ logA,   // (1,)
                        const int*   __restrict__ flag,   // (1,)
                        float*       __restrict__ tiles)  // (B,L,192)
{
    __shared__ __align__(16) float smem[SMU_FLOATS + 2 * SMCOEF_FLOATS];
    float* smU = smem;                                    // staged u0 patches
    float* smA = smem + SMU_FLOATS;                       // coefA region
    float* smC = smem + SMU_FLOATS + SMCOEF_FLOATS;       // coefC region

    const int cx = blockIdx.x;          // j-chunk
    const int i  = blockIdx.y;          // patch row
    const int b  = blockIdx.z;
    const int j0 = cx * CH_J;
    const int npatch = min(CH_J, NWw - j0);
    const int ncols  = min(REG_COLS, Ww - 4 * j0);  // 36 (or 32 at tail)
    const int tid = threadIdx.x;

    // ---- Phase 1: async-stage u0 patches (i-1..i+1) x (j0-1..j0+8) --------
    const size_t u0_base = (size_t)b * LL * EPP;
    for (int task = tid; task < NSLOT * 48; task += TPB) {
        const int slot  = task / 48;          // 0..29
        const int chunk = task % 48;          // 48 x 16B = 192 floats
        const int ip = i - 1 + slot / 10;
        const int jp = j0 - 1 + slot % 10;
        if (ip >= 0 && ip < NHh && jp >= 0 && jp < NWw) {
            const size_t goff = u0_base + (size_t)(ip * NWw + jp) * EPP + chunk * 4;
            const uint32_t lb =
                (uint32_t)(uintptr_t)(&smU[slot * EPP + chunk * 4]);
            async_copy_b128(lb, u0 + goff);
        }
    }
    wait_async_all();
    __syncthreads();

    // ---- Phase 2: per-pixel coefficients into LDS -------------------------
    const float aval = expf(logA[0]);
    const int   fl   = flag[0];
    const int   nent = Cc * 8 * ncols;
    for (int q = tid; q < nent; q += TPB) {
        const int c   = q / (8 * ncols);
        const int rem = q % (8 * ncols);
        const int row = rem / ncols;
        const int col = rem % ncols;
        const int y  = 4 * i  + row;
        const int xx = 4 * j0 + col;

        // gather <=4 covering-patch contributions from LDS
        float S1 = 0.f, Q = 0.f;
        int cnt = 0;
        const int ic = min(y >> 2,  NHh - 1);
        const int jc = min(xx >> 2, NWw - 1);
        #pragma unroll
        for (int a2 = 0; a2 < 2; ++a2) {
            const int ip = ic - a2;
            const int kyv = y - 4 * ip;
            if (ip < 0 || kyv > 7) continue;
            #pragma unroll
            for (int b2 = 0; b2 < 2; ++b2) {
                const int jp = jc - b2;
                const int kxv = xx - 4 * jp;
                if (jp < 0 || kxv > 7) continue;
                const int slot = (ip - (i - 1)) * 10 + (jp - (j0 - 1));
                const float v = smU[slot * EPP + c * 64 + kyv * 8 + kxv];
                S1 += v; Q += v * v; ++cnt;
            }
        }

        const size_t pix = ((size_t)(b * Cc + c) * Hh + y) * Ww + xx;
        const float xv = ximg[pix];
        float cA = 0.f, cC = xv;
        if (fl == 1 && cnt > 1) {                  // N==1 -> u_final == 0
            const float Nf   = (float)cnt;
            const float mean = S1 / Nf;
            float S2 = Q - S1 * mean;              // sum((u0-mean)^2)
            if (S2 < 0.f) S2 = 0.f;
            const float norm = sqrtf(S2);
            const float rv = rimg[pix];
            const float radial = (cnt == 4) ? cbrtf(rv) : rv;  // r^(1/max(N-1,1))
            cA = sqrtf(Nf) * aval * radial / (norm + 1e-6f);
            cC = xv - cA * mean;
        }
        smA[(c * 8 + row) * REG_COLS + col] = cA;
        smC[(c * 8 + row) * REG_COLS + col] = cC;
    }
    __syncthreads();

    // ---- Phase 3: apply affine map, one float4 per thread -----------------
    const int p = tid / 48;               // patch within chunk
    const int s = tid % 48;               // float4 slot within patch
    if (p < npatch) {
        const int e0  = 4 * s;            // 0..188, multiple of 4
        const int c   = e0 >> 6;
        const int rem = e0 & 63;
        const int ky  = rem >> 3;
        const int kx0 = rem & 7;          // 0 or 4
        const int col = 4 * p + kx0;      // region column (16B aligned)
        const int slot = 10 + (p + 1);    // di=1 (patch row i), dj=p+1

        const float4 u4 = *(const float4*)&smU[slot * EPP + e0];
        const float4 a4 = *(const float4*)&smA[(c * 8 + ky) * REG_COLS + col];
        const float4 c4 = *(const float4*)&smC[(c * 8 + ky) * REG_COLS + col];
        float4 o;
        o.x = fmaf(a4.x, u4.x, c4.x);
        o.y = fmaf(a4.y, u4.y, c4.y);
        o.z = fmaf(a4.z, u4.z, c4.z);
        o.w = fmaf(a4.w, u4.w, c4.w);

        const size_t l = (size_t)i * NWw + (j0 + p);
        *(float4*)&tiles[((size_t)b * LL + l) * EPP + e0] = o;
    }
}

// ---------------------------------------------------------------------------
// Ltot: closed form over the fixed count map N (64 pixels N=1, 8064 N=2,
// 254016 N=4).  Ltot = C*( sum(logV) + logA*sum(N-1) + sum(N/2*logN) ).
// ---------------------------------------------------------------------------
__global__ void ltot_kernel(const float* __restrict__ logA,
                            float* __restrict__ out)
{
    const double ln2  = 0.6931471805599453;
    const double lnpi = 1.1447298858494002;
    const double lg25 = 0.28468287047291918;   // lgamma(2.5)
    const double logV4 = 1.5 * lnpi - lg25;    // logV for N=4 (pad_dim=3)
    // logV for N=2 is exactly ln2; N=1 contributes 0.
    const double S_logV  = 8064.0 * ln2 + 254016.0 * logV4;
    const double S_pad   = 770112.0;           // 8064*1 + 254016*3
    const double S_scale = 1024128.0 * ln2;    // 8064*ln2 + 254016*2*ln4
    const double lt = 3.0 * (S_logV + (double)logA[0] * S_pad + S_scale);
    out[0] = (float)lt;
}

// ---------------------------------------------------------------------------
extern "C" void kernel_launch(void* const* d_in, const int* in_sizes, int n_in,
                              void* d_out, int out_size, void* d_ws, size_t ws_size,
                              hipStream_t stream) {
    (void)in_sizes; (void)n_in; (void)out_size; (void)d_ws; (void)ws_size;
    const float* ximg = (const float*)d_in[0];   // x      (B,C,H,W)
    const float* u0   = (const float*)d_in[1];   // u_0    (B,L,192)
    const float* rimg = (const float*)d_in[2];   // r      (B,C,H,W)
    const float* logA = (const float*)d_in[3];   // logA   (1,1)
    const int*   flag = (const int*)d_in[4];     // flag   scalar
    float* out = (float*)d_out;                  // tiles flat + Ltot

    dim3 grid(NCHUNK, NHh, Bb);                  // (16,127,16)
    fused_tiles_kernel<<<grid, TPB, 0, stream>>>(ximg, u0, rimg, logA, flag, out);
    ltot_kernel<<<1, 1, 0, stream>>>(logA, out + (size_t)Bb * LL * EPP);
}